// ChannelSpatialSELayer_90271622627371
// MI455X (gfx1250) — compile-verified
//
#include <hip/hip_runtime.h>
#include <math.h>

typedef float v2f __attribute__((ext_vector_type(2)));
typedef float v4f __attribute__((ext_vector_type(4)));
typedef float v8f __attribute__((ext_vector_type(8)));

#define C_CH    64
#define CR_CH   32
#define BATCH   2
#define SPATIAL 294912   /* 32*96*96 */
#define SVEC    73728    /* SPATIAL/4 */

__device__ __forceinline__ float sigmoidf(float v) {
    return 1.0f / (1.0f + __expf(-v));
}

// ---------------------------------------------------------------------------
// Kernel 1: per-(b,c) plane mean.  128 blocks, one plane each; 1024 threads
// stream the 1.15 MB plane with b128 loads, LDS tree-reduce.
// Side effect: warms the 192 MB L2 with x (151 MB fits).
// ---------------------------------------------------------------------------
__global__ void scse_mean_kernel(const float* __restrict__ x,
                                 float* __restrict__ smean) {
    __shared__ float red[1024];
    const int p = blockIdx.x;                      // b*64 + c
    const v4f* xv = (const v4f*)(x + (size_t)p * SPATIAL);
    float acc = 0.0f;
    for (int i = threadIdx.x; i < SVEC; i += 1024) {
        v4f v = xv[i];
        acc += (v.x + v.y) + (v.z + v.w);
    }
    red[threadIdx.x] = acc;
    __syncthreads();
    for (int off = 512; off > 0; off >>= 1) {
        if (threadIdx.x < off) red[threadIdx.x] += red[threadIdx.x + off];
        __syncthreads();
    }
    if (threadIdx.x == 0) smean[p] = red[0] * (1.0f / (float)SPATIAL);
}

// ---------------------------------------------------------------------------
// Kernel 2: SE MLP via V_WMMA_F32_16X16X4_F32 (exact fp32 GEMM).
// Single wave (32 threads, EXEC all-ones).
//   FC1: h(2x32)      = smean(2x64) @ fc1_w(32x64)^T   -> relu
//   FC2: gate_c(2x64) = h(2x32)     @ fc2_w(64x32)^T   -> sigmoid
// A layout (16x4 f32): lanes 0-15 hold {K0,K1}, lanes 16-31 hold {K2,K3},
// row m = lane&15.  B layout (4x16): mirror with N = lane&15.
// Rows m>=2 of A are duplicates of rows 0/1 (harmless: only feed unused D rows).
// D layout: acc[0] lanes 0-15 = row b=0, acc[1] lanes 0-15 = row b=1.
// ---------------------------------------------------------------------------
__global__ void scse_gates_kernel(const float* __restrict__ smean,
                                  const float* __restrict__ fc1_w,
                                  const float* __restrict__ fc1_b,
                                  const float* __restrict__ fc2_w,
                                  const float* __restrict__ fc2_b,
                                  float* __restrict__ gate_c) {
    __shared__ float hbuf[BATCH][CR_CH];
    const int lane = threadIdx.x;          // 0..31
    const int m    = lane & 15;            // row / column index within tile
    const int mr   = m & 1;                // clamp A-row to {0,1} (in-bounds reads)
    const int k0   = (lane >> 4) << 1;     // 0 for lanes 0-15, 2 for lanes 16-31

    // ---- FC1 ----
    for (int nt = 0; nt < 2; ++nt) {
        v8f acc = {0.f,0.f,0.f,0.f,0.f,0.f,0.f,0.f};
        const int j = nt * 16 + m;         // fc1 output-neuron for this lane's B column
        for (int kb = 0; kb < C_CH; kb += 4) {
            v2f a, bv;
            a.x  = smean[mr * C_CH + kb + k0];
            a.y  = smean[mr * C_CH + kb + k0 + 1];
            bv.x = fc1_w[j * C_CH + kb + k0];
            bv.y = fc1_w[j * C_CH + kb + k0 + 1];
            acc = __builtin_amdgcn_wmma_f32_16x16x4_f32(
                      false, a, false, bv, (short)0, acc, false, false);
        }
        if (lane < 16) {
            const int jj = nt * 16 + lane;
            const float bias = fc1_b[jj];
            hbuf[0][jj] = fmaxf(acc[0] + bias, 0.0f);
            hbuf[1][jj] = fmaxf(acc[1] + bias, 0.0f);
        }
    }
    __syncthreads();

    // ---- FC2 + sigmoid ----
    for (int nt = 0; nt < 4; ++nt) {
        v8f acc = {0.f,0.f,0.f,0.f,0.f,0.f,0.f,0.f};
        const int i = nt * 16 + m;
        for (int kb = 0; kb < CR_CH; kb += 4) {
            v2f a, bv;
            a.x  = hbuf[mr][kb + k0];
            a.y  = hbuf[mr][kb + k0 + 1];
            bv.x = fc2_w[i * CR_CH + kb + k0];
            bv.y = fc2_w[i * CR_CH + kb + k0 + 1];
            acc = __builtin_amdgcn_wmma_f32_16x16x4_f32(
                      false, a, false, bv, (short)0, acc, false, false);
        }
        if (lane < 16) {
            const int ii = nt * 16 + lane;
            const float bias = fc2_b[ii];
            gate_c[ii]        = sigmoidf(acc[0] + bias);
            gate_c[C_CH + ii] = sigmoidf(acc[1] + bias);
        }
    }
}

// ---------------------------------------------------------------------------
// Kernel 3: fused sSE conv + gating + max.  One thread = one float4 of
// spatial positions across all 64 channels.  First channel sweep builds the
// spatial gate; second sweep (hits WGP$/L2 on the re-read) applies both gates.
// Output stored non-temporally so x stays resident in L2.
// ---------------------------------------------------------------------------
__global__ void scse_main_kernel(const float* __restrict__ x,
                                 const float* __restrict__ conv_w,
                                 const float* __restrict__ conv_b,
                                 const float* __restrict__ gate_c,
                                 float* __restrict__ out) {
    const int t = blockIdx.x * blockDim.x + threadIdx.x;   // 0 .. B*SVEC-1
    const int b = t / SVEC;                                // uniform per block
    const int q = t - b * SVEC;
    const v4f* xv = (const v4f*)x;
    v4f*       ov = (v4f*)out;
    const size_t base = (size_t)b * C_CH * SVEC + (size_t)q;

    // channel dot -> spatial squeeze
    v4f acc = {0.f, 0.f, 0.f, 0.f};
    #pragma unroll 16
    for (int c = 0; c < C_CH; ++c) {
        v4f v = xv[base + (size_t)c * SVEC];
        acc += v * conv_w[c];
    }
    const float cb = conv_b[0];
    v4f gs;
    gs.x = sigmoidf(acc.x + cb);
    gs.y = sigmoidf(acc.y + cb);
    gs.z = sigmoidf(acc.z + cb);
    gs.w = sigmoidf(acc.w + cb);

    const float* gcp = gate_c + b * C_CH;
    #pragma unroll 16
    for (int c = 0; c < C_CH; ++c) {
        const size_t idx = base + (size_t)c * SVEC;
        v4f v = xv[idx];
        const float gc = gcp[c];
        v4f o;
        o.x = fmaxf(v.x * gc, v.x * gs.x);
        o.y = fmaxf(v.y * gc, v.y * gs.y);
        o.z = fmaxf(v.z * gc, v.z * gs.z);
        o.w = fmaxf(v.w * gc, v.w * gs.w);
        __builtin_nontemporal_store(o, &ov[idx]);
    }
}

// ---------------------------------------------------------------------------
extern "C" void kernel_launch(void* const* d_in, const int* in_sizes, int n_in,
                              void* d_out, int out_size, void* d_ws, size_t ws_size,
                              hipStream_t stream) {
    const float* x      = (const float*)d_in[0];
    const float* fc1_w  = (const float*)d_in[1];
    const float* fc1_b  = (const float*)d_in[2];
    const float* fc2_w  = (const float*)d_in[3];
    const float* fc2_b  = (const float*)d_in[4];
    const float* conv_w = (const float*)d_in[5];
    const float* conv_b = (const float*)d_in[6];
    float* out = (float*)d_out;

    float* wsf    = (float*)d_ws;
    float* smean  = wsf;            // 128 floats
    float* gate_c = wsf + 128;      // 128 floats

    scse_mean_kernel<<<BATCH * C_CH, 1024, 0, stream>>>(x, smean);
    scse_gates_kernel<<<1, 32, 0, stream>>>(smean, fc1_w, fc1_b, fc2_w, fc2_b, gate_c);
    scse_main_kernel<<<(BATCH * SVEC) / 256, 256, 0, stream>>>(x, conv_w, conv_b, gate_c, out);
}